// GNN_24807731101722
// MI455X (gfx1250) — compile-verified
//
#include <hip/hip_runtime.h>

#define HID 256
#define IN_CH 128

typedef __attribute__((ext_vector_type(2))) float v2f;
typedef __attribute__((ext_vector_type(8))) float v8f;

// ---- gfx1250 async global->LDS path (guarded; falls back to scalar fill) ----
#if defined(__has_builtin)
#if __has_builtin(__builtin_amdgcn_global_load_async_to_lds_b128)
#define HAVE_ASYNC_LDS 1
#endif
#endif
#ifndef HAVE_ASYNC_LDS
#define HAVE_ASYNC_LDS 0
#endif

// Builtin signature (from hipcc diagnostic): arg0 = v4i* (generic),
// arg1 = LDS-side pointer (assume v4i AS(3)*), arg2/arg3 = imm offset / cpol.
typedef int v4i_ __attribute__((vector_size(16)));
typedef __attribute__((address_space(3))) v4i_ lds_v4i;

__device__ __forceinline__ void wait_async_zero() {
#if defined(__has_builtin) && __has_builtin(__builtin_amdgcn_s_wait_asynccnt)
    __builtin_amdgcn_s_wait_asynccnt(0);
#else
    asm volatile("s_wait_asynccnt 0x0" ::: "memory");
#endif
}

// ---------------- degree ----------------
__global__ void deg_kernel(const int* __restrict__ eidx, float* __restrict__ deg, int n_edges) {
    int e = blockIdx.x * blockDim.x + threadIdx.x;
    if (e < n_edges) atomicAdd(&deg[eidx[e]], 1.0f);  // eidx[0..E) = row (target)
}

__global__ void invdeg_kernel(float* __restrict__ deg, int n) {
    int i = blockIdx.x * blockDim.x + threadIdx.x;
    if (i < n) deg[i] = 1.0f / fmaxf(deg[i], 1.0f);
}

// ------------- edge scatter-add (mean-agg numerator) -------------
// one wave per edge; lane covers 4 consecutive floats via float4
template<int C>
__global__ void scatter_kernel(const float* __restrict__ src, const int* __restrict__ eidx,
                               float* __restrict__ agg, int n_edges) {
    int gw   = (int)((blockIdx.x * (size_t)blockDim.x + threadIdx.x) >> 5);
    int lane = threadIdx.x & 31;
    if (gw >= n_edges) return;
    int dst  = eidx[gw];            // row = target
    int srcn = eidx[n_edges + gw];  // col = source
    const float4* s = (const float4*)(src + (size_t)srcn * C);
    float* d = agg + (size_t)dst * C;
#pragma unroll
    for (int base = 0; base < C / 4; base += 32) {
        float4 v = s[base + lane];
        int k = (base + lane) * 4;
        atomicAdd(d + k + 0, v.x);
        atomicAdd(d + k + 1, v.y);
        atomicAdd(d + k + 2, v.z);
        atomicAdd(d + k + 3, v.w);
    }
}

// ------------- fused SAGE layer: relu( (agg*invdeg)@Wl^T + b + hin@Wr^T ) [* p] -------------
// Block = 64 node rows (4 row-tiles of 16), 256 threads = 8 waves.
// Each wave owns 2 column-tiles of 16; per K-step the B fragments (W rows) are
// loaded once and reused across the 4 row-tiles -> 8 WMMAs per B load, and
// 4 independent accumulator chains per wave for pipelining.
// f32 WMMA 16x16x4: A lane<16 -> K={0,1}, lane>=16 -> K={2,3}; B mirrored; D: M=v+8*half.
template<int KD, bool APPLY_P>
__global__ __launch_bounds__(256)
void sage_gemm_kernel(const float* __restrict__ agg, const float* __restrict__ hin,
                      const float* __restrict__ inv_deg,
                      const float* __restrict__ wl, const float* __restrict__ bl,
                      const float* __restrict__ wr, const float* __restrict__ p,
                      float* __restrict__ out, int n_nodes) {
    __shared__ __align__(16) float sA[64][KD + 4];   // +4 pad: 4-bank skew per row
    __shared__ __align__(16) float sH[64][KD + 4];
    const int tid = threadIdx.x;
    const int m0  = blockIdx.x * 64;

    const bool interior = (m0 + 64) <= n_nodes;
#if HAVE_ASYNC_LDS
    if (interior) {
        // pure byte-copy tile fill via async global->LDS (ASYNCcnt), 16B chunks
        const int CHUNKS = 64 * (KD / 4);  // 16B chunks per matrix
        for (int c = tid; c < CHUNKS; c += 256) {
            int r  = c / (KD / 4);
            int kc = c - r * (KD / 4);
            const float* gA = agg + (size_t)(m0 + r) * KD + kc * 4;
            const float* gH = hin + (size_t)(m0 + r) * KD + kc * 4;
            lds_v4i* lA = (lds_v4i*)(unsigned)(uintptr_t)&sA[r][kc * 4];
            lds_v4i* lH = (lds_v4i*)(unsigned)(uintptr_t)&sH[r][kc * 4];
            __builtin_amdgcn_global_load_async_to_lds_b128((v4i_*)gA, lA, 0, 0);
            __builtin_amdgcn_global_load_async_to_lds_b128((v4i_*)gH, lH, 0, 0);
        }
        wait_async_zero();
        __syncthreads();
        // apply mean-normalization row scale in LDS (row-uniform => legal pre-WMMA)
        for (int idx = tid; idx < 64 * KD; idx += 256) {
            int r = idx / KD, k = idx - r * KD;
            sA[r][k] *= inv_deg[m0 + r];
        }
        __syncthreads();
    } else
#endif
    {
        for (int idx = tid; idx < 64 * KD; idx += 256) {
            int r = idx / KD, k = idx - r * KD;
            int row = m0 + r;
            float a = 0.0f, h = 0.0f;
            if (row < n_nodes) {
                a = agg[(size_t)row * KD + k] * inv_deg[row];
                h = hin[(size_t)row * KD + k];
            }
            sA[r][k] = a;
            sH[r][k] = h;
        }
        __syncthreads();
    }

    const int wv = tid >> 5, lane = tid & 31;
    const int half = lane >> 4, mrow = lane & 15;

    for (int ct = wv; ct < HID / 16; ct += 8) {
        const int n0 = ct * 16;
        const float* wlr = wl + (size_t)(n0 + mrow) * KD;  // row (n0+mrow) of Wl
        const float* wrr = wr + (size_t)(n0 + mrow) * KD;
        __builtin_prefetch(wlr, 0, 3);
        __builtin_prefetch(wrr, 0, 3);
        __builtin_prefetch(wlr + KD / 2, 0, 3);
        __builtin_prefetch(wrr + KD / 2, 0, 3);
        v8f acc0 = {0.f,0.f,0.f,0.f,0.f,0.f,0.f,0.f};
        v8f acc1 = acc0, acc2 = acc0, acc3 = acc0;
#pragma unroll 4
        for (int k0 = 0; k0 < KD; k0 += 4) {
            const int ka = k0 + half * 2;
            v2f bL = {wlr[ka], wlr[ka + 1]};
            v2f bR = {wrr[ka], wrr[ka + 1]};
#define SAGE_RT(RT, ACC)                                                                         \
            {                                                                                    \
                v2f aA = {sA[(RT)*16 + mrow][ka], sA[(RT)*16 + mrow][ka + 1]};                   \
                v2f aH = {sH[(RT)*16 + mrow][ka], sH[(RT)*16 + mrow][ka + 1]};                   \
                ACC = __builtin_amdgcn_wmma_f32_16x16x4_f32(false, aA, false, bL, (short)0, ACC, \
                                                            false, false);                       \
                ACC = __builtin_amdgcn_wmma_f32_16x16x4_f32(false, aH, false, bR, (short)0, ACC, \
                                                            false, false);                       \
            }
            SAGE_RT(0, acc0)
            SAGE_RT(1, acc1)
            SAGE_RT(2, acc2)
            SAGE_RT(3, acc3)
#undef SAGE_RT
        }
        const float bias = bl[n0 + mrow];
        const int col = n0 + mrow;
#define SAGE_EPI(RT, ACC)                                                     \
        {                                                                     \
            _Pragma("unroll")                                                 \
            for (int v = 0; v < 8; v++) {                                     \
                int row = m0 + (RT) * 16 + v + half * 8;                      \
                if (row < n_nodes) {                                          \
                    float val = fmaxf(ACC[v] + bias, 0.0f);                   \
                    if (APPLY_P) val *= p[row];                               \
                    out[(size_t)row * HID + col] = val;                       \
                }                                                             \
            }                                                                 \
        }
        SAGE_EPI(0, acc0)
        SAGE_EPI(1, acc1)
        SAGE_EPI(2, acc2)
        SAGE_EPI(3, acc3)
#undef SAGE_EPI
    }
}

// ------------- p-weighted global mean pool (numerator + counts) -------------
__global__ void pool_kernel(const float* __restrict__ h2, const float* __restrict__ p,
                            const int* __restrict__ batch, float* __restrict__ gsum,
                            float* __restrict__ gcnt, int n_nodes) {
    int node = (int)((blockIdx.x * (size_t)blockDim.x + threadIdx.x) >> 5);
    int lane = threadIdx.x & 31;
    if (node >= n_nodes) return;
    int g = batch[node];
    float pv = p[node];
    if (lane == 0) atomicAdd(&gcnt[g], 1.0f);
    const float4* s = (const float4*)(h2 + (size_t)node * HID);
    float* d = gsum + (size_t)g * HID;
#pragma unroll
    for (int base = 0; base < HID / 4; base += 32) {
        float4 v = s[base + lane];
        int k = (base + lane) * 4;
        atomicAdd(d + k + 0, v.x * pv);
        atomicAdd(d + k + 1, v.y * pv);
        atomicAdd(d + k + 2, v.z * pv);
        atomicAdd(d + k + 3, v.w * pv);
    }
}

// ------------- readout classifier: [h2[root] || gmean] @ lin_w^T + lin_b -------------
__global__ void classifier_kernel(const float* __restrict__ h2, const int* __restrict__ root_ptr,
                                  const float* __restrict__ gsum, const float* __restrict__ gcnt,
                                  const float* __restrict__ lin_w, const float* __restrict__ lin_b,
                                  float* __restrict__ out) {
    __shared__ float sf[2 * HID];
    const int g = blockIdx.x;
    const int tid = threadIdx.x;  // 64 threads; one output channel each
    const int root = root_ptr[g];
    const float inv_cnt = 1.0f / fmaxf(gcnt[g], 1.0f);
    for (int i = tid; i < HID; i += 64) {
        sf[i]       = h2[(size_t)root * HID + i];
        sf[HID + i] = gsum[(size_t)g * HID + i] * inv_cnt;
    }
    __syncthreads();
    float acc = lin_b[tid];
    const float* wrow = lin_w + (size_t)tid * (2 * HID);
#pragma unroll 8
    for (int cidx = 0; cidx < 2 * HID; cidx++) acc += sf[cidx] * wrow[cidx];
    out[(size_t)g * 64 + tid] = acc;
}

extern "C" void kernel_launch(void* const* d_in, const int* in_sizes, int n_in,
                              void* d_out, int out_size, void* d_ws, size_t ws_size,
                              hipStream_t stream) {
    const float* x        = (const float*)d_in[0];
    const int*   eidx     = (const int*)d_in[1];
    const int*   root_ptr = (const int*)d_in[2];
    const float* p        = (const float*)d_in[3];
    const int*   batch    = (const int*)d_in[4];
    // d_in[5] = group_ptr: unused by the reference
    const float* w1_l = (const float*)d_in[6];
    const float* b1_l = (const float*)d_in[7];
    const float* w1_r = (const float*)d_in[8];
    const float* w2_l = (const float*)d_in[9];
    const float* b2_l = (const float*)d_in[10];
    const float* w2_r = (const float*)d_in[11];
    const float* lin_w = (const float*)d_in[12];
    const float* lin_b = (const float*)d_in[13];
    float* out = (float*)d_out;

    const int n_nodes  = in_sizes[0] / IN_CH;
    const int n_edges  = in_sizes[1] / 2;
    const int n_graphs = in_sizes[2];

    // workspace carve-up (floats, 1KB-aligned chunks)
    size_t off = 0;
    float* ws = (float*)d_ws;
    auto carve = [&](size_t nf) { float* q = ws + off; off += (nf + 255) & ~(size_t)255; return q; };
    float* inv_deg = carve((size_t)n_nodes);
    float* agg     = carve((size_t)n_nodes * HID);
    float* h1p     = carve((size_t)n_nodes * HID);
    float* h2      = carve((size_t)n_nodes * HID);
    float* gsum    = carve((size_t)n_graphs * HID);
    float* gcnt    = carve((size_t)n_graphs);

    const int edge_waves_grid = (n_edges + 7) / 8;      // 8 waves (edges) per 256-thread block
    const int node_tile_grid  = (n_nodes + 63) / 64;    // 64 node rows per block

    // degree -> 1/max(deg,1)
    (void)hipMemsetAsync(inv_deg, 0, (size_t)n_nodes * sizeof(float), stream);
    deg_kernel<<<(n_edges + 255) / 256, 256, 0, stream>>>(eidx, inv_deg, n_edges);
    invdeg_kernel<<<(n_nodes + 255) / 256, 256, 0, stream>>>(inv_deg, n_nodes);

    // layer 1: agg1 = scatter(x); h1p = relu(agg1*inv@W1l^T + b1 + x@W1r^T) * p
    (void)hipMemsetAsync(agg, 0, (size_t)n_nodes * IN_CH * sizeof(float), stream);
    scatter_kernel<IN_CH><<<edge_waves_grid, 256, 0, stream>>>(x, eidx, agg, n_edges);
    sage_gemm_kernel<IN_CH, true><<<node_tile_grid, 256, 0, stream>>>(
        agg, x, inv_deg, w1_l, b1_l, w1_r, p, h1p, n_nodes);

    // layer 2: agg2 = scatter(h1p); h2 = relu(agg2*inv@W2l^T + b2 + h1p@W2r^T)
    (void)hipMemsetAsync(agg, 0, (size_t)n_nodes * HID * sizeof(float), stream);
    scatter_kernel<HID><<<edge_waves_grid, 256, 0, stream>>>(h1p, eidx, agg, n_edges);
    sage_gemm_kernel<HID, false><<<node_tile_grid, 256, 0, stream>>>(
        agg, h1p, inv_deg, w2_l, b2_l, w2_r, p, h2, n_nodes);

    // readout
    (void)hipMemsetAsync(gsum, 0, (size_t)n_graphs * HID * sizeof(float), stream);
    (void)hipMemsetAsync(gcnt, 0, (size_t)n_graphs * sizeof(float), stream);
    pool_kernel<<<(n_nodes + 7) / 8, 256, 0, stream>>>(h2, p, batch, gsum, gcnt, n_nodes);
    classifier_kernel<<<n_graphs, 64, 0, stream>>>(h2, root_ptr, gsum, gcnt, lin_w, lin_b, out);
}